// OmegaRelGraphConv_23862838297346
// MI455X (gfx1250) — compile-verified
//
#include <hip/hip_runtime.h>
#include <hip/hip_bf16.h>

typedef float v2f __attribute__((ext_vector_type(2)));
typedef float v4f __attribute__((ext_vector_type(4)));
typedef float v8f __attribute__((ext_vector_type(8)));

#define D_FEAT 128
#define LDS_STRIDE 132   // 128 + 4 pad: rows stay 16B-aligned, B-fragment ds_load_b64 conflict-free
#define RRELU_SLOPE 0.2291666666666667f
#define GEMM_THREADS 512
#define GEMM_WAVES 16

// ---------------------------------------------------------------------------
// grid-stride zero
__global__ void zero_kernel(float* __restrict__ p, size_t n) {
    size_t i = (size_t)blockIdx.x * blockDim.x + threadIdx.x;
    size_t stride = (size_t)gridDim.x * blockDim.x;
    for (; i < n; i += stride) p[i] = 0.0f;
}

// ---------------------------------------------------------------------------
// one-time pass: deg[dst] += 1, esum[dst] += edge_feats[e]   (one wave per edge)
__global__ void edge_pre_kernel(const float* __restrict__ ef,
                                const int* __restrict__ dst,
                                float* __restrict__ esum,
                                float* __restrict__ deg,
                                int n_edges) {
    int t = blockIdx.x * blockDim.x + threadIdx.x;
    int e = t >> 5;
    int lane = t & 31;
    if (e >= n_edges) return;
    int d = dst[e];
    if (lane == 0) atomicAdd(&deg[d], 1.0f);
    // edge features are streamed exactly once in the whole problem -> NT hint,
    // keep L2 free for the node/agg tables (25.6 MB each, fully L2-resident).
    v4f v = __builtin_nontemporal_load((const v4f*)(ef + (size_t)e * D_FEAT + lane * 4));
    float* base = esum + (size_t)d * D_FEAT + lane * 4;
    atomicAdd(base + 0, v[0]);
    atomicAdd(base + 1, v[1]);
    atomicAdd(base + 2, v[2]);
    atomicAdd(base + 3, v[3]);
}

// per-layer pass: agg[dst] += h[src]   (one wave per edge; agg pre-seeded with esum)
__global__ void edge_gather_kernel(const float* __restrict__ h,
                                   const int* __restrict__ src,
                                   const int* __restrict__ dst,
                                   float* __restrict__ agg,
                                   int n_edges) {
    int t = blockIdx.x * blockDim.x + threadIdx.x;
    int e = t >> 5;
    int lane = t & 31;
    if (e >= n_edges) return;
    int s = src[e];
    int d = dst[e];
    v4f v = *(const v4f*)(h + (size_t)s * D_FEAT + lane * 4);
    float* base = agg + (size_t)d * D_FEAT + lane * 4;
    atomicAdd(base + 0, v[0]);
    atomicAdd(base + 1, v[1]);
    atomicAdd(base + 2, v[2]);
    atomicAdd(base + 3, v[3]);
}

// ---------------------------------------------------------------------------
// node update:  out = leakyrelu( (agg/max(deg,1)) @ Wn^T
//                                + [deg>0]·h @ Ws^T + [deg==0]·h @ Wi^T )
// One wave per 16-node tile, 16 waves (512 thr) per block.  All three weight
// matrices live in LDS simultaneously (3 * 128 * 132 * 4 = 202,752 B < 320 KB),
// so a single fused k-loop accumulates all three products into one C tile.
// A layout: lane l -> (M = l&15, K = 4t + 2*(l>>4) + {0,1})
// B layout: lane l -> (N = l&15, K = 4t + 2*(l>>4) + {0,1}),  B[k][n] = W[n][k]
__global__ void __launch_bounds__(GEMM_THREADS, 1)
node_gemm_kernel(const float* __restrict__ agg,
                 const float* __restrict__ h,
                 const float* __restrict__ deg,
                 const float* __restrict__ Wn,
                 const float* __restrict__ Ws,
                 const float* __restrict__ Wi,
                 float* __restrict__ out,
                 int n_tiles, int n_nodes) {
    extern __shared__ float sW[];                 // [3][128][LDS_STRIDE]
    const int tid  = threadIdx.x;
    const int lane = tid & 31;
    const int wave = tid >> 5;
    const int tile = blockIdx.x * GEMM_WAVES + wave;
    const bool active = (tile < n_tiles);

    // cooperative load of Wn|Ws|Wi into padded LDS (b128 stores, coalesced)
    for (int i = tid; i < 3 * 128 * 32; i += GEMM_THREADS) {
        const int m  = i >> 5;                    // matrix*128 + row, 0..383
        const int c4 = i & 31;
        const float* Wsrc = (m < 128) ? Wn : ((m < 256) ? Ws : Wi);
        const int row = m & 127;
        ((v4f*)(sW + m * LDS_STRIDE))[c4] = ((const v4f*)(Wsrc + row * D_FEAT))[c4];
    }
    __syncthreads();

    const int r     = lane & 15;                  // row-in-tile (M) / col-in-jtile (N)
    const int khalf = (lane >> 4) * 2;            // 0 or 2

    // Unconditional loads: clamp node for tail waves, mask via scale factors.
    int node = tile * 16 + r;
    if (node >= n_nodes) node = n_nodes - 1;
    const float dg = deg[node];
    const float sN = active ? (1.0f / fmaxf(dg, 1.0f)) : 0.0f;  // mean scale
    const float sS = (active && dg >  0.0f) ? 1.0f : 0.0f;      // self mask
    const float sI = (active && dg == 0.0f) ? 1.0f : 0.0f;      // isolated mask

    const float* aggRow = agg + (size_t)node * D_FEAT + khalf;
    const float* hRow   = h   + (size_t)node * D_FEAT + khalf;
    const float* sWn = sW;
    const float* sWs = sW + 128 * LDS_STRIDE;
    const float* sWi = sW + 256 * LDS_STRIDE;
    const int boff = r * LDS_STRIDE + khalf;

    v8f acc[8] = {};                              // 16x128 fp32 output tile

#pragma unroll
    for (int t = 0; t < 32; ++t) {
        const int k0 = t * 4;
        v2f av = *(const v2f*)(aggRow + k0);
        v2f hv = *(const v2f*)(hRow + k0);
        v2f an = {av[0] * sN, av[1] * sN};
        v2f as = {hv[0] * sS, hv[1] * sS};
        v2f ai = {hv[0] * sI, hv[1] * sI};
#pragma unroll
        for (int j = 0; j < 8; ++j) {
            v2f b = *(const v2f*)(sWn + j * 16 * LDS_STRIDE + boff + k0);
            acc[j] = __builtin_amdgcn_wmma_f32_16x16x4_f32(
                false, an, false, b, (short)0, acc[j], false, false);
        }
#pragma unroll
        for (int j = 0; j < 8; ++j) {
            v2f b = *(const v2f*)(sWs + j * 16 * LDS_STRIDE + boff + k0);
            acc[j] = __builtin_amdgcn_wmma_f32_16x16x4_f32(
                false, as, false, b, (short)0, acc[j], false, false);
        }
#pragma unroll
        for (int j = 0; j < 8; ++j) {
            v2f b = *(const v2f*)(sWi + j * 16 * LDS_STRIDE + boff + k0);
            acc[j] = __builtin_amdgcn_wmma_f32_16x16x4_f32(
                false, ai, false, b, (short)0, acc[j], false, false);
        }
    }

    if (active) {
        const int mofs = (lane < 16) ? 0 : 8;     // C layout: VGPR rr = rows rr / rr+8
#pragma unroll
        for (int j = 0; j < 8; ++j) {
#pragma unroll
            for (int rr = 0; rr < 8; ++rr) {
                float v = acc[j][rr];
                v = (v >= 0.0f) ? v : v * RRELU_SLOPE;
                const size_t row = (size_t)tile * 16 + rr + mofs;
                out[row * D_FEAT + j * 16 + r] = v;
            }
        }
    }
}

// ---------------------------------------------------------------------------
extern "C" void kernel_launch(void* const* d_in, const int* in_sizes, int n_in,
                              void* d_out, int out_size, void* d_ws, size_t ws_size,
                              hipStream_t stream) {
    const float* node_feats = (const float*)d_in[0];
    const float* edge_feats = (const float*)d_in[1];
    const int*   src        = (const int*)d_in[2];
    const int*   dst        = (const int*)d_in[3];
    const float* Wn0 = (const float*)d_in[4];
    const float* Ws0 = (const float*)d_in[5];
    const float* Wi0 = (const float*)d_in[6];
    const float* Wn1 = (const float*)d_in[7];
    const float* Ws1 = (const float*)d_in[8];
    const float* Wi1 = (const float*)d_in[9];
    float* out = (float*)d_out;

    const int N = in_sizes[0] / D_FEAT;           // 50000
    const int E = in_sizes[2];                    // 800000
    const size_t ND = (size_t)N * D_FEAT;

    // workspace layout (needs (N + 3*N*D)*4 ≈ 77 MB)
    float* w    = (float*)d_ws;
    float* deg  = w;                              // N (contiguous with esum for one zero pass)
    float* esum = w + N;                          // N*D
    float* agg  = esum + ND;                      // N*D
    float* h1   = agg + ND;                       // N*D

    const int n_tiles     = (N + 15) / 16;        // 3125
    const int gemm_blocks = (n_tiles + GEMM_WAVES - 1) / GEMM_WAVES;
    const size_t lds_bytes = (size_t)3 * 128 * LDS_STRIDE * sizeof(float);  // 202752
    const int edge_blocks = (int)(((size_t)E * 32 + 255) / 256);

    // deg = 0, esum = 0 (contiguous)
    zero_kernel<<<2048, 256, 0, stream>>>(deg, (size_t)N + ND);

    // one-time: degree + edge-feature segment sum (reused by both layers)
    edge_pre_kernel<<<edge_blocks, 256, 0, stream>>>(edge_feats, dst, esum, deg, E);

    // ---- layer 0 ----
    hipMemcpyAsync(agg, esum, ND * sizeof(float), hipMemcpyDeviceToDevice, stream);
    edge_gather_kernel<<<edge_blocks, 256, 0, stream>>>(node_feats, src, dst, agg, E);
    node_gemm_kernel<<<gemm_blocks, GEMM_THREADS, lds_bytes, stream>>>(
        agg, node_feats, deg, Wn0, Ws0, Wi0, h1, n_tiles, N);

    // ---- layer 1 ----
    hipMemcpyAsync(agg, esum, ND * sizeof(float), hipMemcpyDeviceToDevice, stream);
    edge_gather_kernel<<<edge_blocks, 256, 0, stream>>>(h1, src, dst, agg, E);
    node_gemm_kernel<<<gemm_blocks, GEMM_THREADS, lds_bytes, stream>>>(
        agg, h1, deg, Wn1, Ws1, Wi1, out, n_tiles, N);
}